// DifferentiableEulerSolver_88742614269980
// MI455X (gfx1250) — compile-verified
//
#include <hip/hip_runtime.h>
#include <cstdint>

// Problem geometry (fixed by the reference)
#define N        128
#define NN       (N * N)
#define NNN      (N * NN)
#define SUBSTEPS 10
#define NSTEPS   40            // MAX_DAYS * SUBSTEPS
#define MICRO_DT 0.1f

// Tiling
#define TX 32
#define TY 8
#define BZ 32
#define ZCH (N / BZ)           // 4 z-chunks
#define PW (TX + 2)            // 34
#define PH (TY + 2)            // 10
#define PLANE (PW * PH)        // 340 floats
#define PLANE_B (PLANE * 4)    // bytes per plane slot
#define NTHREADS (TX * TY)     // 256 threads = 8 wave32s

// ---- CDNA5 async global->LDS copy (gfx1250) -------------------------------
// lds_off is the LDS destination byte address; tracked by ASYNCcnt, per-lane under EXEC.
__device__ __forceinline__ void async_copy_f32(uint32_t lds_off, const float* gptr) {
  asm volatile("global_load_async_to_lds_b32 %0, %1, off"
               :: "v"(lds_off), "v"(gptr)
               : "memory");
}

__device__ __forceinline__ void wait_async_all() {
  asm volatile("s_wait_asynccnt 0x0" ::: "memory");
}
// ---------------------------------------------------------------------------

__global__ __launch_bounds__(NTHREADS)
void fk_step(const float* __restrict__ uin,
             const float* __restrict__ Dm,
             const float* __restrict__ rm,
             const int*   __restrict__ dt_days,
             float*       __restrict__ uout,
             int day, int doClip)
{
  // 4 rotating z-plane slots with +1 halo in x and y.
  __shared__ float pl[4][PLANE];

  const int tx  = threadIdx.x;
  const int ty  = threadIdx.y;
  const int tid = ty * TX + tx;

  const int x0 = blockIdx.x * TX;
  const int y0 = blockIdx.y * TY;
  const int b  = blockIdx.z / ZCH;
  const int z0 = (blockIdx.z % ZCH) * BZ;

  const int   base  = b * NNN;                      // < 2^22, 32-bit safe
  const float scale = (day < dt_days[b]) ? MICRO_DT : 0.0f;

  // ---- hoisted plane-load setup: each thread owns elements tid, tid+256 ---
  const int i0 = tid;
  const int i1 = tid + NTHREADS;
  const int lx0 = (i0 % PW) - 1, ly0 = (i0 / PW) - 1;
  const int lx1 = (i1 % PW) - 1, ly1 = (i1 / PW) - 1;
  const int gx0 = x0 + lx0, gy0 = y0 + ly0;
  const int gx1 = x0 + lx1, gy1 = y0 + ly1;
  const bool inb0 = (gx0 >= 0) & (gx0 < N) & (gy0 >= 0) & (gy0 < N);
  const bool inb1 = (i1 < PLANE) &
                    ((gx1 >= 0) & (gx1 < N) & (gy1 >= 0) & (gy1 < N));
  const int ro0 = gy0 * N + gx0;                    // in-plane offset (elems)
  const int ro1 = gy1 * N + gx1;
  const uint32_t lo0 = (uint32_t)(uintptr_t)&pl[0][i0];   // slot-0 LDS byte addr
  const uint32_t lo1 = (uint32_t)(uintptr_t)&pl[0][i1 < PLANE ? i1 : 0];

  // Zero all slots once: Dirichlet zeros for halo cells that are never
  // loaded (x/y edges map to the same slot cells each rotation; z=-1 plane
  // is never written).
  for (int i = tid; i < 4 * PLANE; i += NTHREADS)
    ((float*)pl)[i] = 0.0f;
  __syncthreads();

  // issue async loads for one plane (zp guaranteed in [0,N))
  auto issue_plane = [&](int zp, const float* pb /* = uin+base+zp*NN */) {
    const uint32_t so = (uint32_t)((zp + 1) & 3) * PLANE_B;
    if (inb0) async_copy_f32(lo0 + so, pb + ro0);
    if (inb1) async_copy_f32(lo1 + so, pb + ro1);
  };

  // ---- prologue: planes z0-1, z0, z0+1 ------------------------------------
  const float* pb = uin + base + (size_t)z0 * NN;
  if (z0 > 0) issue_plane(z0 - 1, pb - NN);         // z=-1 stays zero
  issue_plane(z0, pb);
  issue_plane(z0 + 1, pb + NN);
  wait_async_all();
  __syncthreads();

  const int cy = (ty + 1) * PW + (tx + 1);

  // running pointers (advance by one plane per z)
  const int row = (y0 + ty) * N + (x0 + tx);
  const float* pnext = pb + 2 * NN;                 // plane z0+2
  const float* Dp = Dm   + base + (size_t)z0 * NN + row;
  const float* rp = rm   + base + (size_t)z0 * NN + row;
  float*       op = uout + base + (size_t)z0 * NN + row;

  // z-column register rotation: zm (z-1), c (z); zq (z+1) read per iter.
  float zm = pl[(z0) & 3][cy];                      // slot(z0-1)
  float c  = pl[(z0 + 1) & 3][cy];                  // slot(z0)

  for (int z = z0; z < z0 + BZ; ++z) {
    // Overlap: fetch plane z+2 into the unused 4th slot while computing z.
    const int zp = z + 2;
    if (zp <= z0 + BZ) {
      if (zp < N) {
        issue_plane(zp, pnext);
      } else {                                      // z=128: clear stale data
        const uint32_t slot = (uint32_t)((zp + 1) & 3);
        for (int i = tid; i < PLANE; i += NTHREADS) pl[slot][i] = 0.0f;
      }
      pnext += NN;
    }

    const int s1 = (z + 1) & 3;                     // plane z
    const float xm = pl[s1][cy - 1];
    const float xp = pl[s1][cy + 1];
    const float ym = pl[s1][cy - PW];
    const float yp = pl[s1][cy + PW];
    const float zq = pl[(z + 2) & 3][cy];           // plane z+1

    const float lap = (xm + xp + ym + yp + zm + zq) - 6.0f * c;
    const float du  = (*Dp) * lap + (*rp) * c * (1.0f - c);
    float r = c + scale * du;
    if (doClip) r = fminf(fmaxf(r, 0.0f), 1.0f);
    *op = r;

    zm = c;
    c  = zq;
    Dp += NN; rp += NN; op += NN;

    // Plane z+2 resident + all waves done reading slot (z-1) before it is
    // overwritten next iteration.
    wait_async_all();
    __syncthreads();
  }
}

extern "C" void kernel_launch(void* const* d_in, const int* in_sizes, int n_in,
                              void* d_out, int out_size, void* d_ws, size_t ws_size,
                              hipStream_t stream) {
  (void)n_in; (void)out_size; (void)ws_size;

  const float* u0 = (const float*)d_in[0];
  const float* Dm = (const float*)d_in[1];
  const float* rm = (const float*)d_in[2];
  const int*   dt = (const int*)d_in[3];

  float* out = (float*)d_out;
  float* ws  = (float*)d_ws;              // needs B*128^3 floats = 16 MB

  const int B = in_sizes[3];              // batch (2)

  dim3 block(TX, TY, 1);
  dim3 grid(N / TX, N / TY, ZCH * B);     // 4 x 16 x 8 = 512 blocks

  const float* src = u0;
  for (int step = 0; step < NSTEPS; ++step) {
    const int day  = step / SUBSTEPS;
    float*    dst  = (step & 1) ? out : ws;   // step 39 (last, odd) -> d_out
    const int clip = (step == NSTEPS - 1);
    fk_step<<<grid, block, 0, stream>>>(src, Dm, rm, dt, dst, day, clip);
    src = dst;
  }
}